// PromptDetectionLoss_78048145703361
// MI455X (gfx1250) — compile-verified
//
#include <hip/hip_runtime.h>
#include <hip/hip_bf16.h>
#include <math.h>

#define N_ANCH 172032
#define C_CLS  256
#define REGMAX 16
#define M_BOX  64
#define GB 672    /* N_ANCH / 256 exactly */
#define GC 2048
#define GD 2688   /* N_ANCH*4 / 256 exactly */

#define LOG2E 1.4426950408889634f
#define LN2   0.6931471805599453f

typedef __attribute__((ext_vector_type(2))) float v2f;
typedef __attribute__((ext_vector_type(8))) float v8f;

// ---------------------------------------------------------------------------
// Wave-wide sum via V_WMMA_F32_16X16X4_F32 (B = ones => row sums of A).
// a={x,0}: D[m,*] = x[m] + x[m+16]; lanes<16 hold rows 0-7, lanes>=16 rows
// 8-15; sum of own 8 D regs + xor-16 exchange = full 32-lane sum on all lanes.
// Requires EXEC all-ones (call sites divergence-free).
// ---------------------------------------------------------------------------
__device__ __forceinline__ float wave_sum_wmma(float x) {
  v2f a; a[0] = x;    a[1] = 0.0f;
  v2f b; b[0] = 1.0f; b[1] = 1.0f;
  v8f c = {};
  v8f d = __builtin_amdgcn_wmma_f32_16x16x4_f32(false, a, false, b,
                                                (short)0, c, false, false);
  float s = d[0] + d[1] + d[2] + d[3] + d[4] + d[5] + d[6] + d[7];
  s += __shfl_xor(s, 16, 32);
  return s;
}

template <int Q>
__device__ __forceinline__ void emit_block_partials(const float* q,
                                                    float* out_part,
                                                    float* sred /* 8*Q */) {
  const int wave = threadIdx.x >> 5;
  const int lane = threadIdx.x & 31;
#pragma unroll
  for (int i = 0; i < Q; ++i) {
    float s = wave_sum_wmma(q[i]);
    if (lane == 0) sred[wave * Q + i] = s;
  }
  __syncthreads();
  if (threadIdx.x == 0) {
#pragma unroll
    for (int i = 0; i < Q; ++i) {
      float t = 0.f;
      for (int w = 0; w < 8; ++w) t += sred[w * Q + i];
      out_part[i] = t;
    }
  }
}

// ---------------------------------------------------------------------------
// Kernel A: per-box "exists inside&center anchor" bitmask.
// Per-box constants precomputed once per block into LDS.
// ---------------------------------------------------------------------------
__global__ void pdl_kA(const float* __restrict__ ap,
                       const float* __restrict__ gtb,
                       unsigned int* __restrict__ hc) {
  __shared__ float sx1[M_BOX], sy1[M_BOX], sx2[M_BOX], sy2[M_BOX];
  __shared__ float scx[M_BOX], scy[M_BOX], srx[M_BOX], sry[M_BOX];
  __shared__ unsigned int sOr[2];
  const int t = threadIdx.x;
  if (t < M_BOX) {
    const float4 bx = ((const float4*)gtb)[t];
    sx1[t] = bx.x; sy1[t] = bx.y; sx2[t] = bx.z; sy2[t] = bx.w;
    scx[t] = (bx.x + bx.z) * 0.5f;
    scy[t] = (bx.y + bx.w) * 0.5f;
    srx[t] = __builtin_amdgcn_rcpf(fmaxf((bx.z - bx.x) * 0.5f, 1.f));
    sry[t] = __builtin_amdgcn_rcpf(fmaxf((bx.w - bx.y) * 0.5f, 1.f));
  }
  if (t < 2) sOr[t] = 0u;
  __syncthreads();

  const int n = blockIdx.x * 256 + t;
  const float ax = ap[2 * n], ay = ap[2 * n + 1];
  unsigned int m0 = 0u, m1 = 0u;
#pragma unroll 4
  for (int m = 0; m < M_BOX; ++m) {
    const bool inside = (ax >= sx1[m]) & (ax <= sx2[m]) &
                        (ay >= sy1[m]) & (ay <= sy2[m]);
    const float dx = (ax - scx[m]) * srx[m];
    const float dy = (ay - scy[m]) * sry[m];
    const bool center = fmaxf(fabsf(dx), fabsf(dy)) <= 0.75f;
    if (inside && center) {
      if (m < 32) m0 |= (1u << m); else m1 |= (1u << (m - 32));
    }
  }
  if (m0) atomicOr(&sOr[0], m0);
  if (m1) atomicOr(&sOr[1], m1);
  __syncthreads();
  if (t == 0) {
    if (sOr[0]) atomicOr(&hc[0], sOr[0]);
    if (sOr[1]) atomicOr(&hc[1], sOr[1]);
  }
}

// ---------------------------------------------------------------------------
// Kernel B: region_w / heat per anchor; fused obj-BCE and oversize penalty.
// Partials: {sum(region+heat), sum(region), sum(obj_bce), sum(penalty)}
// ---------------------------------------------------------------------------
__global__ void pdl_kB(const float* __restrict__ ap,
                       const float* __restrict__ gtb,
                       const float* __restrict__ bw,
                       const float* __restrict__ pboxes,
                       const float* __restrict__ obj,
                       const unsigned int* __restrict__ hc,
                       float* __restrict__ region_w,
                       float* __restrict__ heat_out,
                       float* __restrict__ partB) {
  __shared__ float sx1[M_BOX], sy1[M_BOX], sx2[M_BOX], sy2[M_BOX];
  __shared__ float scx[M_BOX], scy[M_BOX], srx[M_BOX], sry[M_BOX];
  __shared__ float sbw[M_BOX];
  __shared__ int   shas[M_BOX];
  __shared__ float sred[8 * 4];
  const int t = threadIdx.x;
  if (t < M_BOX) {
    const float4 bx = ((const float4*)gtb)[t];
    sx1[t] = bx.x; sy1[t] = bx.y; sx2[t] = bx.z; sy2[t] = bx.w;
    scx[t] = (bx.x + bx.z) * 0.5f;
    scy[t] = (bx.y + bx.w) * 0.5f;
    srx[t] = __builtin_amdgcn_rcpf(fmaxf((bx.z - bx.x) * 0.5f, 1.f));
    sry[t] = __builtin_amdgcn_rcpf(fmaxf((bx.w - bx.y) * 0.5f, 1.f));
    sbw[t] = bw[t];
    shas[t] = (int)((hc[t >> 5] >> (t & 31)) & 1u);
  }
  __syncthreads();

  const int n = blockIdx.x * 256 + t;
  const float ax = ap[2 * n], ay = ap[2 * n + 1];

  float rw = 0.f, hmax = 0.f;
#pragma unroll 4
  for (int m = 0; m < M_BOX; ++m) {
    const bool inside = (ax >= sx1[m]) & (ax <= sx2[m]) &
                        (ay >= sy1[m]) & (ay <= sy2[m]);
    const float dx = (ax - scx[m]) * srx[m];
    const float dy = (ay - scy[m]) * sry[m];
    const bool center = fmaxf(fabsf(dx), fabsf(dy)) <= 0.75f;
    const bool cand = inside & (center | (shas[m] == 0));
    rw = fmaxf(rw, cand ? sbw[m] : 0.f);
    // sigma=2: exp(-0.125*(dx^2+dy^2)) = exp2(-0.125*log2e*(dx^2+dy^2))
    const float hv = inside
        ? __builtin_amdgcn_exp2f(-0.125f * LOG2E * (dx * dx + dy * dy))
        : 0.f;
    hmax = fmaxf(hmax, hv);
  }
  hmax = fminf(fmaxf(hmax, 0.f), 1.f);
  region_w[n] = rw;
  heat_out[n] = hmax;

  // objectness BCE against heat (softplus via hw exp2/log2)
  const float l = obj[n];
  const float t2 = __builtin_amdgcn_exp2f(-LOG2E * fabsf(l));
  const float bce = fmaxf(l, 0.f) - l * hmax +
                    LN2 * __builtin_amdgcn_logf(1.0f + t2);

  // oversize box penalty on pred_boxes[n]
  const float4 pbx = ((const float4*)pboxes)[n];
  const float x1c = fminf(fmaxf(pbx.x, 0.f), 1024.f);
  const float y1c = fminf(fmaxf(pbx.y, 0.f), 1024.f);
  const float x2c = fminf(fmaxf(pbx.z, 0.f), 1024.f);
  const float y2c = fminf(fmaxf(pbx.w, 0.f), 1024.f);
  const float bwd = fmaxf(x2c - x1c, 0.f);
  const float bht = fmaxf(y2c - y1c, 0.f);
  const float ar = bwd * bht * (1.0f / (1024.f * 1024.f));
  const float edge = ((x1c <= 1.f ? 1.f : 0.f) + (y1c <= 1.f ? 1.f : 0.f) +
                      (x2c >= 1023.f ? 1.f : 0.f) + (y2c >= 1023.f ? 1.f : 0.f)) * 0.25f;
  const float pen = fmaxf(ar - 0.6f, 0.f) * (1.f + edge);

  float q[4] = {rw + hmax, rw, bce, pen};
  emit_block_partials<4>(q, &partB[blockIdx.x * 4], sred);
}

// ---------------------------------------------------------------------------
// Kernel C: the 352 MB streaming pass. One wave per row (32 lanes x 8 ch).
// Hardware transcendentals: 3 TRANS + ~14 VALU per element.
// Partials: {sum(vfl * w), sum(relu(rowmax - 8))}
// ---------------------------------------------------------------------------
__global__ void pdl_kC(const float* __restrict__ logits,
                       const float* __restrict__ tscores,
                       const float* __restrict__ region_w,
                       const float* __restrict__ heat,
                       float* __restrict__ partC) {
  __shared__ float sred[8 * 2];
  const int lane = threadIdx.x & 31;
  const int wave = threadIdx.x >> 5;
  const int gw = (blockIdx.x << 3) + wave;
  const int nw = GC << 3;

  float accv = 0.f, accm = 0.f;
  for (int n = gw; n < N_ANCH; n += nw) {
    const float w = region_w[n] + heat[n];
    const size_t base = (size_t)n * C_CLS + (lane << 3);
    const float4* lp = (const float4*)(logits + base);
    const float4* tp = (const float4*)(tscores + base);
    if (n + nw < N_ANCH) {   // L2 prefetch of next row (global_prefetch_b8)
      __builtin_prefetch(logits + base + (size_t)nw * C_CLS, 0, 1);
      __builtin_prefetch(tscores + base + (size_t)nw * C_CLS, 0, 1);
    }
    const float4 l0 = lp[0], l1 = lp[1];
    const float4 t0 = tp[0], t1 = tp[1];
    const float la[8] = {l0.x, l0.y, l0.z, l0.w, l1.x, l1.y, l1.z, l1.w};
    const float ta[8] = {t0.x, t0.y, t0.z, t0.w, t1.x, t1.y, t1.z, t1.w};
    float rsum = 0.f, rmax = -3.4e38f;
#pragma unroll
    for (int i = 0; i < 8; ++i) {
      const float l = la[i], tt = ta[i];
      const float t2 = __builtin_amdgcn_exp2f(-LOG2E * fabsf(l)); // e^{-|l|}
      const float r1 = __builtin_amdgcn_rcpf(1.0f + t2);          // 1/(1+t2)
      const float p = (l >= 0.f) ? r1 : (t2 * r1);                // sigmoid
      const float sp = LN2 * __builtin_amdgcn_logf(1.0f + t2);    // log1p(t2)
      const float bce = fmaxf(l, 0.f) - l * tt + sp;
      const float wt = fmaf(0.75f * p * p, 1.f - tt, tt);
      rsum += bce * wt;
      rmax = fmaxf(rmax, l);
    }
    accv += rsum * w;
#pragma unroll
    for (int off = 1; off < 32; off <<= 1)
      rmax = fmaxf(rmax, __shfl_xor(rmax, off, 32));
    if (lane == 0) accm += fmaxf(rmax - 8.0f, 0.f);
  }
  float q[2] = {accv, accm};
  emit_block_partials<2>(q, &partC[blockIdx.x * 2], sred);
}

// ---------------------------------------------------------------------------
// Kernel D: DFL loss. Thread per (n, side): 16-bin log-softmax (hw exp2/log2).
// ---------------------------------------------------------------------------
__global__ void pdl_kD(const float* __restrict__ pred_dist,
                       const float* __restrict__ dfl_target,
                       const float* __restrict__ region_w,
                       float* __restrict__ partD) {
  __shared__ float sred[8 * 1];
  const int tid0 = blockIdx.x * 256 + threadIdx.x;
  const int stride = GD * 256;
  float acc = 0.f;
  for (int id = tid0; id < N_ANCH * 4; id += stride) {
    const int n = id >> 2;
    const float rw = region_w[n];
    const float4* pp = (const float4*)(pred_dist + (size_t)id * REGMAX);
    const float4 a = pp[0], b = pp[1], c = pp[2], d = pp[3];
    const float p[16] = {a.x, a.y, a.z, a.w, b.x, b.y, b.z, b.w,
                         c.x, c.y, c.z, c.w, d.x, d.y, d.z, d.w};
    const float t = dfl_target[id];
    int tl = (int)t; tl = min(max(tl, 0), REGMAX - 1);
    const int tr = min(tl + 1, REGMAX - 1);
    const float wl = (float)tr - t;
    const float wr = 1.f - wl;
    float mx = p[0];
#pragma unroll
    for (int i = 1; i < 16; ++i) mx = fmaxf(mx, p[i]);
    float se = 0.f;
#pragma unroll
    for (int i = 0; i < 16; ++i)
      se += __builtin_amdgcn_exp2f(LOG2E * (p[i] - mx));
    const float lse = mx + LN2 * __builtin_amdgcn_logf(se);
    float ptl = 0.f, ptr_ = 0.f;
#pragma unroll
    for (int i = 0; i < 16; ++i) {
      if (i == tl) ptl = p[i];
      if (i == tr) ptr_ = p[i];
    }
    acc += ((lse - ptl) * wl + (lse - ptr_) * wr) * rw;
  }
  float q[1] = {acc};
  emit_block_partials<1>(q, &partD[blockIdx.x], sred);
}

// ---------------------------------------------------------------------------
// Kernel E: deterministic fixed-order combine of all block partials.
// ---------------------------------------------------------------------------
__device__ __forceinline__ float block_sum_fixed(float v, float* sh) {
  const int t = threadIdx.x;
  sh[t] = v;
  __syncthreads();
  for (int s = 128; s > 0; s >>= 1) {
    if (t < s) sh[t] += sh[t + s];
    __syncthreads();
  }
  const float r = sh[0];
  __syncthreads();
  return r;
}

__global__ void pdl_kE(const float* __restrict__ pB,
                       const float* __restrict__ pC,
                       const float* __restrict__ pD,
                       float* __restrict__ out) {
  __shared__ float sh[256];
  const int t = threadIdx.x;
  float sums[7];
#pragma unroll
  for (int q = 0; q < 4; ++q) {
    float a = 0.f;
    for (int b = t; b < GB; b += 256) a += pB[b * 4 + q];
    sums[q] = block_sum_fixed(a, sh);
  }
#pragma unroll
  for (int q = 0; q < 2; ++q) {
    float a = 0.f;
    for (int b = t; b < GC; b += 256) a += pC[b * 2 + q];
    sums[4 + q] = block_sum_fixed(a, sh);
  }
  {
    float a = 0.f;
    for (int b = t; b < GD; b += 256) a += pD[b];
    sums[6] = block_sum_fixed(a, sh);
  }
  if (t == 0) {
    const float sum_w = sums[0], sum_rw = sums[1], sum_obj = sums[2];
    const float sum_pen = sums[3], sum_vfl = sums[4], sum_margin = sums[5];
    const float sum_dfl = sums[6];
    const float invN = 1.0f / (float)N_ANCH;
    const float cls_loss = sum_vfl / fmaxf(sum_w, 1e-6f);
    const float obj_loss = sum_obj * invN;
    const float dfl_loss_v = sum_dfl / fmaxf(sum_rw, 1e-6f);
    const float pen = sum_pen * invN;
    const float mar = sum_margin * invN;
    out[0] = cls_loss + obj_loss + dfl_loss_v + 0.1f * pen + 0.05f * mar;
  }
}

// ---------------------------------------------------------------------------
extern "C" void kernel_launch(void* const* d_in, const int* in_sizes, int n_in,
                              void* d_out, int out_size, void* d_ws, size_t ws_size,
                              hipStream_t stream) {
  const float* ap  = (const float*)d_in[0];  // anchor_points [N,2]
  const float* cls = (const float*)d_in[1];  // cls_logits    [N,C]
  const float* ts  = (const float*)d_in[2];  // target_scores [N,C]
  const float* pd  = (const float*)d_in[3];  // pred_dist     [N,4,16]
  const float* dt  = (const float*)d_in[4];  // dfl_target    [N,4]
  const float* gtb = (const float*)d_in[5];  // gt_boxes      [M,4]
  const float* bw  = (const float*)d_in[6];  // box_weights   [M]
  const float* pb  = (const float*)d_in[7];  // pred_boxes    [N,4]
  const float* obj = (const float*)d_in[8];  // obj_logits    [N]
  float* out = (float*)d_out;

  float* ws = (float*)d_ws;
  float* region_w = ws;                       // N floats
  float* heat     = ws + N_ANCH;              // N floats
  float* partB    = ws + 2 * N_ANCH;          // GB*4
  float* partC    = partB + GB * 4;           // GC*2
  float* partD    = partC + GC * 2;           // GD
  unsigned int* hc = (unsigned int*)(partD + GD);  // 2 u32

  hipMemsetAsync(hc, 0, 2 * sizeof(unsigned int), stream);
  pdl_kA<<<GB, 256, 0, stream>>>(ap, gtb, hc);
  pdl_kB<<<GB, 256, 0, stream>>>(ap, gtb, bw, pb, obj, hc, region_w, heat, partB);
  pdl_kC<<<GC, 256, 0, stream>>>(cls, ts, region_w, heat, partC);
  pdl_kD<<<GD, 256, 0, stream>>>(pd, dt, region_w, partD);
  pdl_kE<<<1, 256, 0, stream>>>(partB, partC, partD, out);
}